// TripletLoss_65000035058015
// MI455X (gfx1250) — compile-verified
//
#include <hip/hip_runtime.h>

typedef float v2f __attribute__((ext_vector_type(2)));
typedef float v8f __attribute__((ext_vector_type(8)));

#define NB     64      // batch per term
#define CDIM   256     // channels
#define KDIM   196     // k spatial (always 14x14)
#define KPAD   209     // LDS row pitch (odd -> spread banks)
#define NWAVES 8
#define NTERMS 14

__global__ __launch_bounds__(256) void triplet_main_kernel(
    const float* __restrict__ gl_sa,   // (2*64, 256, 14*14)
    const float* __restrict__ lo_sa,   // (6*64, 256, 10*10)
    const float* __restrict__ gl_pr,   // (2*64, 256, 14*14)
    const float* __restrict__ coords,  // (8, 64, 4)
    float* __restrict__ partial)       // (14*64) per-block hinge
{
    extern __shared__ float sdata[];                 // NWAVES * 16 * KPAD
    __shared__ float s_ckx[KDIM];
    __shared__ float s_cky[KDIM];
    __shared__ float s_pn[NWAVES], s_pd[NWAVES], s_ng[NWAVES];

    const int tid  = threadIdx.x;
    const int lane = tid & 31;
    const int wave = tid >> 5;
    const int li   = lane & 15;
    const int h    = lane >> 4;

    const int blk = blockIdx.x;
    const int t = blk / NB;          // term 0..13
    const int n = blk % NB;
    const int j = t / 7;             // which gl_proj split
    const int s = t % 7;             // 0 = gl-gl term, 1..6 = lo terms

    const float* kf = gl_pr  + (size_t)(j * NB + n) * CDIM * KDIM;
    const float* ck = coords + (size_t)(j * NB + n) * 4;

    const float* qf;
    const float* cq;
    int Q, Hq;
    if (s == 0) {
        int i = 1 - j;
        qf = gl_sa  + (size_t)(i * NB + n) * CDIM * 196;
        cq = coords + (size_t)(i * NB + n) * 4;
        Q = 196; Hq = 14;
    } else {
        int i = s - 1;
        qf = lo_sa  + (size_t)(i * NB + n) * CDIM * 100;
        cq = coords + (size_t)((2 + i) * NB + n) * 4;
        Q = 100; Hq = 10;
    }
    const int Wq = Hq;

    // ---- geometry ----
    const float q0 = cq[0], q1 = cq[1], q2 = cq[2], q3 = cq[3];
    const float k0 = ck[0], k1 = ck[1], k2 = ck[2], k3 = ck[3];
    const float bwq = (q2 - q0) / (float)Wq;
    const float bhq = (q3 - q1) / (float)Hq;
    const float bwk = (k2 - k0) / 14.0f;
    const float bhk = (k3 - k1) / 14.0f;
    const float qd = sqrtf(bwq * bwq + bhq * bhq);
    const float kd = sqrtf(bwk * bwk + bhk * bhk);
    const float md = fmaxf(qd, kd);
    float thr2 = 0.7f * md;          // dist < 0.7  <=>  dx^2+dy^2 < (0.7*md)^2
    thr2 *= thr2;

    for (int p = tid; p < KDIM; p += blockDim.x) {
        int ky = p / 14, kx = p % 14;
        s_ckx[p] = ((float)kx + 0.5f) * bwk + k0;
        s_cky[p] = ((float)ky + 0.5f) * bhk + k1;
    }
    __syncthreads();

    float pnum = 0.0f, pden = 0.0f, gneg = 0.0f;

    const int qTiles = (Q + 15) >> 4;
    const int nIter  = (qTiles + NWAVES - 1) / NWAVES;
    float* strip = sdata + wave * 16 * KPAD;

    for (int it = 0; it < nIter; ++it) {
        const int qt = wave + it * NWAVES;
        if (qt < qTiles) {
            // ---- 16 x 196 strip of S = qf^T * kf via f32 WMMA ----
            int m = qt * 16 + li; if (m > Q - 1) m = Q - 1;      // clamp edge rows
            const float* aB = qf + (size_t)(2 * h) * Q + m;      // A[m, c+2h+v]

            // k-tiles 0..11 in groups of 4: 1 A-frag feeds 4 WMMAs
            for (int ktg = 0; ktg < 3; ++ktg) {
                const int colBase = ktg * 64 + li;               // max 191 < 196: no clamp
                const float* bB = kf + (size_t)(2 * h) * KDIM + colBase;
                v8f a0 = {0.f,0.f,0.f,0.f,0.f,0.f,0.f,0.f};
                v8f a1 = a0, a2 = a0, a3 = a0;
#pragma unroll 2
                for (int c = 0; c < CDIM; c += 4) {
                    v2f a;
                    a.x = aB[(size_t)c * Q];
                    a.y = aB[(size_t)(c + 1) * Q];
                    const float* bc = bB + (size_t)c * KDIM;
                    v2f b0, b1, b2, b3;
                    b0.x = bc[0];        b0.y = bc[KDIM];
                    b1.x = bc[16];       b1.y = bc[KDIM + 16];
                    b2.x = bc[32];       b2.y = bc[KDIM + 32];
                    b3.x = bc[48];       b3.y = bc[KDIM + 48];
                    a0 = __builtin_amdgcn_wmma_f32_16x16x4_f32(false, a, false, b0, (short)0, a0, false, false);
                    a1 = __builtin_amdgcn_wmma_f32_16x16x4_f32(false, a, false, b1, (short)0, a1, false, false);
                    a2 = __builtin_amdgcn_wmma_f32_16x16x4_f32(false, a, false, b2, (short)0, a2, false, false);
                    a3 = __builtin_amdgcn_wmma_f32_16x16x4_f32(false, a, false, b3, (short)0, a3, false, false);
                }
                float* srow = strip + 8 * h * KPAD + ktg * 64 + li;
#pragma unroll
                for (int r = 0; r < 8; ++r) {
                    srow[r * KPAD +  0] = a0[r];
                    srow[r * KPAD + 16] = a1[r];
                    srow[r * KPAD + 32] = a2[r];
                    srow[r * KPAD + 48] = a3[r];
                }
            }

            // tail k-tile 12 (cols 192..207, clamped)
            {
                int col = 192 + li; if (col > KDIM - 1) col = KDIM - 1;
                const float* bB = kf + (size_t)(2 * h) * KDIM + col;
                v8f acc = {0.f,0.f,0.f,0.f,0.f,0.f,0.f,0.f};
#pragma unroll 4
                for (int c = 0; c < CDIM; c += 4) {
                    v2f a, b;
                    a.x = aB[(size_t)c * Q];
                    a.y = aB[(size_t)(c + 1) * Q];
                    b.x = bB[(size_t)c * KDIM];
                    b.y = bB[(size_t)(c + 1) * KDIM];
                    acc = __builtin_amdgcn_wmma_f32_16x16x4_f32(false, a, false, b, (short)0, acc, false, false);
                }
#pragma unroll
                for (int r = 0; r < 8; ++r)
                    strip[(r + 8 * h) * KPAD + 192 + li] = acc[r];
            }
        }
        // same-wave LDS store -> scan ordering (strips are per-wave private)
        asm volatile("s_wait_dscnt 0" ::: "memory");

        // ---- fused mask / positives / top-10-smallest scan ----
        const int qt2 = wave + it * NWAVES;
        const int row = qt2 * 16 + li;
        if (h == 0 && qt2 < qTiles && row < Q) {
            const int yq = row / Wq, xq = row % Wq;
            const float cqx = ((float)xq + 0.5f) * bwq + q0;
            const float cqy = ((float)yq + 0.5f) * bhq + q1;
            float t10[10];
#pragma unroll
            for (int x = 0; x < 10; ++x) t10[x] = __builtin_inff();
            for (int k = 0; k < KDIM; ++k) {
                const float sv = strip[li * KPAD + k];
                const float logit = -2.0f * sv;
                const float dx = cqx - s_ckx[k];
                const float dy = cqy - s_cky[k];
                const bool pm = (dx * dx + dy * dy) < thr2;
                if (pm) {
                    pnum += logit;
                    pden += 1.0f;
                } else if (logit < t10[9]) {
                    t10[9] = logit;          // insert & re-sort ascending
#pragma unroll
                    for (int x = 9; x >= 1; --x) {
                        const float lo2 = fminf(t10[x - 1], t10[x]);
                        const float hi2 = fmaxf(t10[x - 1], t10[x]);
                        t10[x - 1] = lo2;
                        t10[x]     = hi2;
                    }
                }
            }
            float rn = 0.0f;                 // drop smallest, sum 2nd..10th
#pragma unroll
            for (int x = 1; x < 10; ++x) rn += t10[x];
            gneg += rn;
        }
    }

    // ---- wave then block reduction ----
#pragma unroll
    for (int off = 16; off >= 1; off >>= 1) {
        pnum += __shfl_xor(pnum, off, 32);
        pden += __shfl_xor(pden, off, 32);
        gneg += __shfl_xor(gneg, off, 32);
    }
    if (lane == 0) { s_pn[wave] = pnum; s_pd[wave] = pden; s_ng[wave] = gneg; }
    __syncthreads();
    if (tid == 0) {
        float PN = 0.f, PD = 0.f, NG = 0.f;
        for (int w = 0; w < NWAVES; ++w) { PN += s_pn[w]; PD += s_pd[w]; NG += s_ng[w]; }
        const float pos = PN / (PD + 1e-6f);
        const float neg = (NG / 9.0f) / (float)Q;
        const float hinge = fmaxf(0.0f, -(neg - 2.0f * pos) + 100.0f);
        partial[blk] = hinge;
    }
}

// Deterministic final reduction: sum 896 partials, scale by 1/(N * n_terms).
__global__ __launch_bounds__(256) void triplet_reduce_kernel(
    const float* __restrict__ partial, float* __restrict__ out)
{
    __shared__ float red[NWAVES];
    const int tid  = threadIdx.x;
    const int lane = tid & 31;
    const int wave = tid >> 5;
    float acc = 0.0f;
    for (int i = tid; i < NTERMS * NB; i += 256) acc += partial[i];
#pragma unroll
    for (int off = 16; off >= 1; off >>= 1) acc += __shfl_xor(acc, off, 32);
    if (lane == 0) red[wave] = acc;
    __syncthreads();
    if (tid == 0) {
        float s = 0.0f;
        for (int w = 0; w < NWAVES; ++w) s += red[w];
        out[0] = s * (1.0f / ((float)NB * (float)NTERMS));
    }
}

extern "C" void kernel_launch(void* const* d_in, const int* in_sizes, int n_in,
                              void* d_out, int out_size, void* d_ws, size_t ws_size,
                              hipStream_t stream) {
    (void)in_sizes; (void)n_in; (void)out_size; (void)ws_size;
    const float* gl_sa  = (const float*)d_in[0];
    const float* lo_sa  = (const float*)d_in[1];
    const float* gl_pr  = (const float*)d_in[2];
    const float* coords = (const float*)d_in[3];
    float* out     = (float*)d_out;
    float* partial = (float*)d_ws;             // 14*64 floats of scratch

    const size_t shmem = (size_t)NWAVES * 16 * KPAD * sizeof(float); // ~107 KB of 320 KB WGP LDS
    (void)hipFuncSetAttribute((const void*)triplet_main_kernel,
                              hipFuncAttributeMaxDynamicSharedMemorySize, (int)shmem);

    triplet_main_kernel<<<NTERMS * NB, 256, shmem, stream>>>(
        gl_sa, lo_sa, gl_pr, coords, partial);
    triplet_reduce_kernel<<<1, 256, 0, stream>>>(partial, out);
}